// HRGNN_54082228191469
// MI455X (gfx1250) — compile-verified
//
#include <hip/hip_runtime.h>
#include <math.h>

#define D 128
#define NEG_SLOPE 0.01f

typedef float v2f __attribute__((ext_vector_type(2)));
typedef float v4f __attribute__((ext_vector_type(4)));
typedef float v8f __attribute__((ext_vector_type(8)));

// ---------- ordered-uint encoding for float atomic max ----------
__device__ __forceinline__ unsigned f2ord(float f) {
    unsigned b = __float_as_uint(f);
    return (b & 0x80000000u) ? ~b : (b | 0x80000000u);
}
__device__ __forceinline__ float ord2f(unsigned u) {
    return (u & 0x80000000u) ? __uint_as_float(u & 0x7FFFFFFFu)
                             : __uint_as_float(~u);
}

// ---------- K1: z = h @ fc_w^T  (WMMA f32 16x16x4) ----------
// grid = N/16 blocks, 256 threads (8 waves). Wave w owns output cols [16w,16w+16).
__global__ __launch_bounds__(256) void k_gemm_z(const float* __restrict__ h,
                                                const float* __restrict__ fc_w,
                                                float* __restrict__ z) {
    __shared__ float atile[16 * 130];        // 16 rows of h, padded stride
    __shared__ float btile[8][64 * 16];      // per-wave B chunk: [k][jj], k in 64-chunk
    const int t = threadIdx.x;
    const int w = t >> 5, lane = t & 31;
    const int m = lane & 15;                  // row (A) / col (B) within tile
    const int kb = (lane >> 4) << 1;          // lane-half K base (0 or 2)
    const int row0 = blockIdx.x * 16;

    // stage A tile (16x128), coalesced
    for (int i = 0; i < 8; ++i) {
        int idx = t + i * 256;
        int r = idx >> 7, c = idx & 127;
        atile[r * 130 + c] = h[(size_t)(row0 + r) * D + c];
    }

    v8f acc = {};
    for (int kc = 0; kc < D; kc += 64) {
        __syncthreads();
        // stage B chunk: B[k][j] = fc_w[j*D + k] for j in wave's 16 cols
        for (int jj = 0; jj < 16; ++jj) {
            int row = w * 16 + jj;
            for (int k = lane; k < 64; k += 32)
                btile[w][k * 16 + jj] = fc_w[(size_t)row * D + kc + k];
        }
        __syncthreads();
        for (int k0 = 0; k0 < 64; k0 += 4) {
            v2f a, b;
            a.x = atile[m * 130 + kc + k0 + kb];
            a.y = atile[m * 130 + kc + k0 + kb + 1];
            b.x = btile[w][(k0 + kb) * 16 + m];
            b.y = btile[w][(k0 + kb + 1) * 16 + m];
            acc = __builtin_amdgcn_wmma_f32_16x16x4_f32(
                false, a, false, b, (short)0, acc, false, false);
        }
    }
    // C/D layout: VGPR v -> row v (lanes 0-15) or v+8 (lanes 16-31), col = lane&15
    const int rsel = (lane >> 4) << 3;
    const int col = w * 16 + m;
    for (int v = 0; v < 8; ++v)
        z[(size_t)(row0 + rsel + v) * D + col] = acc[v];
}

// scalar tail for leftover rows (not hit when N % 16 == 0)
__global__ void k_z_tail(const float* __restrict__ h, const float* __restrict__ fc_w,
                         float* __restrict__ z, int row_start, int N) {
    int row = row_start + blockIdx.x;
    int col = threadIdx.x;
    if (row >= N) return;
    float s = 0.f;
    for (int k = 0; k < D; ++k) s += h[(size_t)row * D + k] * fc_w[(size_t)col * D + k];
    z[(size_t)row * D + col] = s;
}

// ---------- K2: per-node attention partial scores (wave/node, b128 loads) ----------
__global__ void k_scores(const float* __restrict__ z, const float* __restrict__ attn_w,
                         float* __restrict__ s0, float* __restrict__ s1, int N) {
    int gw = (int)((blockIdx.x * (size_t)blockDim.x + threadIdx.x) >> 5);
    int lane = threadIdx.x & 31;
    if (gw >= N) return;
    const v4f* zp  = (const v4f*)(z + (size_t)gw * D);
    const v4f* a0p = (const v4f*)(attn_w);
    const v4f* a1p = (const v4f*)(attn_w + D);
    v4f zv = zp[lane];
    v4f w0 = a0p[lane];
    v4f w1 = a1p[lane];
    float a0 = zv.x * w0.x + zv.y * w0.y + zv.z * w0.z + zv.w * w0.w;
    float a1 = zv.x * w1.x + zv.y * w1.y + zv.z * w1.z + zv.w * w1.w;
    for (int off = 16; off > 0; off >>= 1) {
        a0 += __shfl_xor(a0, off, 32);
        a1 += __shfl_xor(a1, off, 32);
    }
    if (lane == 0) { s0[gw] = a0; s1[gw] = a1; }
}

// ---------- K3: v_r = aw2 @ fc_r_w ----------
__global__ void k_vr(const float* __restrict__ attn_w, const float* __restrict__ fc_r_w,
                     float* __restrict__ vr) {
    int k = threadIdx.x;
    float s = 0.f;
    for (int j = 0; j < D; ++j) s += attn_w[2 * D + j] * fc_r_w[(size_t)j * D + k];
    vr[k] = s;
}

// ---------- K4: edge scores + leaky relu + segment max (wave per edge) ----------
// r_h is streamed once (410 MB) -> non-temporal b128 loads so it does not evict
// the L2-resident node-sized buffers.
__global__ void k_edge_score(const float* __restrict__ r_h, const float* __restrict__ vr,
                             const float* __restrict__ s0, const float* __restrict__ s1,
                             const int* __restrict__ src, const int* __restrict__ dst,
                             float* __restrict__ ebuf, unsigned* __restrict__ menc, int E) {
    int e = (int)((blockIdx.x * (size_t)blockDim.x + threadIdx.x) >> 5);
    int lane = threadIdx.x & 31;
    if (e >= E) return;
    const v4f* rp = (const v4f*)(r_h + (size_t)e * D);
    const v4f* vp = (const v4f*)vr;
    v4f rv = __builtin_nontemporal_load(rp + lane);
    v4f vv = vp[lane];
    float s = rv.x * vv.x + rv.y * vv.y + rv.z * vv.z + rv.w * vv.w;
    for (int off = 16; off > 0; off >>= 1) s += __shfl_xor(s, off, 32);
    if (lane == 0) {
        float val = s + s0[src[e]] + s1[dst[e]];
        val = (val >= 0.f) ? val : NEG_SLOPE * val;
        ebuf[e] = val;
        atomicMax(&menc[dst[e]], f2ord(val));
    }
}

// ---------- K5: ex = exp(e - m[dst]); den[dst] += ex ----------
__global__ void k_edge_exp(float* __restrict__ ebuf, const int* __restrict__ dst,
                           const unsigned* __restrict__ menc, float* __restrict__ den, int E) {
    int e = (int)(blockIdx.x * (size_t)blockDim.x + threadIdx.x);
    if (e >= E) return;
    int d = dst[e];
    float ex = expf(ebuf[e] - ord2f(menc[d]));
    ebuf[e] = ex;
    atomicAdd(&den[d], ex);
}

// ---------- K6: scatter alpha*r_h -> u[dst], alpha*z[src] -> aggz[dst] ----------
// Second (final) streaming pass over r_h -> non-temporal. Atomic targets and the
// z gather (25.6 MB) stay hot in the 192 MB L2.
__global__ void k_scatter(const float* __restrict__ r_h, const float* __restrict__ z,
                          const float* __restrict__ ebuf, const float* __restrict__ den,
                          const int* __restrict__ src, const int* __restrict__ dst,
                          float* __restrict__ u, float* __restrict__ aggz, int E) {
    int e = (int)((blockIdx.x * (size_t)blockDim.x + threadIdx.x) >> 5);
    int lane = threadIdx.x & 31;
    if (e >= E) return;
    int d = dst[e], sn = src[e];
    float alpha = ebuf[e] / den[d];
    const v4f* rp = (const v4f*)(r_h + (size_t)e * D);
    const v4f* zp = (const v4f*)(z + (size_t)sn * D);
    v4f rv = __builtin_nontemporal_load(rp + lane);
    v4f zv = zp[lane];
    int c = lane * 4;
    float* up = u + (size_t)d * D + c;
    float* ap = aggz + (size_t)d * D + c;
    atomicAdd(up + 0, alpha * rv.x);
    atomicAdd(up + 1, alpha * rv.y);
    atomicAdd(up + 2, alpha * rv.z);
    atomicAdd(up + 3, alpha * rv.w);
    atomicAdd(ap + 0, alpha * zv.x);
    atomicAdd(ap + 1, alpha * zv.y);
    atomicAdd(ap + 2, alpha * zv.z);
    atomicAdd(ap + 3, alpha * zv.w);
}

// ---------- K7: out = relu(aggz + u @ fc_r_w^T + z @ loop_w)  (fused WMMA) ----------
__global__ __launch_bounds__(256) void k_gemm_out(const float* __restrict__ u,
                                                  const float* __restrict__ zmat,
                                                  const float* __restrict__ aggz,
                                                  const float* __restrict__ fc_r_w,
                                                  const float* __restrict__ loop_w,
                                                  float* __restrict__ out) {
    __shared__ float au[16 * 130];
    __shared__ float az[16 * 130];
    __shared__ float btile[8][64 * 16];
    const int t = threadIdx.x;
    const int w = t >> 5, lane = t & 31;
    const int m = lane & 15;
    const int kb = (lane >> 4) << 1;
    const int row0 = blockIdx.x * 16;

    for (int i = 0; i < 8; ++i) {
        int idx = t + i * 256;
        int r = idx >> 7, c = idx & 127;
        au[r * 130 + c] = u[(size_t)(row0 + r) * D + c];
        az[r * 130 + c] = zmat[(size_t)(row0 + r) * D + c];
    }

    v8f acc = {};
    // pass 1: u @ fc_r_w^T  (B[k][j] = fc_r_w[j*D+k])
    for (int kc = 0; kc < D; kc += 64) {
        __syncthreads();
        for (int jj = 0; jj < 16; ++jj) {
            int row = w * 16 + jj;
            for (int k = lane; k < 64; k += 32)
                btile[w][k * 16 + jj] = fc_r_w[(size_t)row * D + kc + k];
        }
        __syncthreads();
        for (int k0 = 0; k0 < 64; k0 += 4) {
            v2f a, b;
            a.x = au[m * 130 + kc + k0 + kb];
            a.y = au[m * 130 + kc + k0 + kb + 1];
            b.x = btile[w][(k0 + kb) * 16 + m];
            b.y = btile[w][(k0 + kb + 1) * 16 + m];
            acc = __builtin_amdgcn_wmma_f32_16x16x4_f32(
                false, a, false, b, (short)0, acc, false, false);
        }
    }
    // pass 2: z @ loop_w  (B[k][j] = loop_w[k*D+j], NOT transposed)
    for (int kc = 0; kc < D; kc += 64) {
        __syncthreads();
        for (int idx = lane; idx < 64 * 16; idx += 32) {
            int k = idx >> 4, jj = idx & 15;
            btile[w][idx] = loop_w[(size_t)(kc + k) * D + w * 16 + jj];
        }
        __syncthreads();
        for (int k0 = 0; k0 < 64; k0 += 4) {
            v2f a, b;
            a.x = az[m * 130 + kc + k0 + kb];
            a.y = az[m * 130 + kc + k0 + kb + 1];
            b.x = btile[w][(k0 + kb) * 16 + m];
            b.y = btile[w][(k0 + kb + 1) * 16 + m];
            acc = __builtin_amdgcn_wmma_f32_16x16x4_f32(
                false, a, false, b, (short)0, acc, false, false);
        }
    }
    const int rsel = (lane >> 4) << 3;
    const int col = w * 16 + m;
    for (int v = 0; v < 8; ++v) {
        size_t idx = (size_t)(row0 + rsel + v) * D + col;
        float val = acc[v] + aggz[idx];
        __builtin_nontemporal_store(fmaxf(val, 0.f), out + idx);  // write-once output
    }
}

__global__ void k_out_tail(const float* __restrict__ u, const float* __restrict__ zmat,
                           const float* __restrict__ aggz, const float* __restrict__ fc_r_w,
                           const float* __restrict__ loop_w, float* __restrict__ out,
                           int row_start, int N) {
    int row = row_start + blockIdx.x;
    int col = threadIdx.x;
    if (row >= N) return;
    float s = aggz[(size_t)row * D + col];
    for (int k = 0; k < D; ++k) {
        s += u[(size_t)row * D + k] * fc_r_w[(size_t)col * D + k];
        s += zmat[(size_t)row * D + k] * loop_w[(size_t)k * D + col];
    }
    out[(size_t)row * D + col] = fmaxf(s, 0.f);
}

extern "C" void kernel_launch(void* const* d_in, const int* in_sizes, int n_in,
                              void* d_out, int out_size, void* d_ws, size_t ws_size,
                              hipStream_t stream) {
    const float* h      = (const float*)d_in[0];
    const float* r_h    = (const float*)d_in[1];
    const float* fc_w   = (const float*)d_in[2];
    const float* fc_r_w = (const float*)d_in[3];
    const float* attn_w = (const float*)d_in[4];
    const float* loop_w = (const float*)d_in[5];
    const int*   src    = (const int*)d_in[6];
    const int*   dst    = (const int*)d_in[7];

    const int N = in_sizes[0] / D;
    const int E = in_sizes[1] / D;

    // workspace layout (floats): [aggz N*D][u N*D][den N][menc N] | [z N*D][s0 N][s1 N][vr D][ebuf E]
    float* ws    = (float*)d_ws;
    float* aggz  = ws;
    float* uacc  = ws + (size_t)N * D;
    float* den   = ws + 2 * (size_t)N * D;
    unsigned* menc = (unsigned*)(ws + 2 * (size_t)N * D + N);
    float* z     = ws + 2 * (size_t)N * D + 2 * (size_t)N;
    float* s0    = z + (size_t)N * D;
    float* s1    = s0 + N;
    float* vr    = s1 + N;
    float* ebuf  = vr + D;

    // zero the accumulated region every call (atomics accumulate into it).
    // menc==0 decodes below any real float -> acts as -inf for the segment max.
    (void)hipMemsetAsync(aggz, 0, (2 * (size_t)N * D + 2 * (size_t)N) * sizeof(float), stream);

    int nblk = N / 16, nrem = N % 16;
    k_gemm_z<<<nblk, 256, 0, stream>>>(h, fc_w, z);
    if (nrem) k_z_tail<<<nrem, D, 0, stream>>>(h, fc_w, z, nblk * 16, N);

    k_scores<<<(N + 7) / 8, 256, 0, stream>>>(z, attn_w, s0, s1, N);
    k_vr<<<1, D, 0, stream>>>(attn_w, fc_r_w, vr);

    k_edge_score<<<(E + 7) / 8, 256, 0, stream>>>(r_h, vr, s0, s1, src, dst, ebuf, menc, E);
    k_edge_exp<<<(E + 255) / 256, 256, 0, stream>>>(ebuf, dst, menc, den, E);
    k_scatter<<<(E + 7) / 8, 256, 0, stream>>>(r_h, z, ebuf, den, src, dst, uacc, aggz, E);

    k_gemm_out<<<nblk, 256, 0, stream>>>(uacc, z, aggz, fc_r_w, loop_w, (float*)d_out);
    if (nrem) k_out_tail<<<nrem, D, 0, stream>>>(uacc, z, aggz, fc_r_w, loop_w,
                                                 (float*)d_out, nblk * 16, N);
}